// QuantizedConv1d_52896817217664
// MI455X (gfx1250) — compile-verified
//
#include <hip/hip_runtime.h>
#include <stdint.h>

typedef __attribute__((ext_vector_type(2))) float v2f;
typedef __attribute__((ext_vector_type(4))) float f32x4;
typedef __attribute__((ext_vector_type(8))) float v8f;
typedef __attribute__((ext_vector_type(4))) int   v4i;

typedef __attribute__((address_space(1))) v4i gv4i_t;   // global int4
typedef __attribute__((address_space(3))) v4i lv4i_t;   // LDS int4

#define CIN     64
#define COUT    64
#define NTILE   128   // L columns per tile
#define TPB     4     // tiles per block (double-buffered pipeline)
#define WSTRIDE 66    // padded W row stride (floats)
#define XSTRIDE 136   // padded x-tile row stride (floats): +2 rows -> +16 banks
#define QOFF    32768.0f

#if defined(__has_builtin)
#if __has_builtin(__builtin_amdgcn_global_load_async_to_lds_b128)
#define HAVE_ASYNC_LDS 1
#endif
#endif

__device__ __forceinline__ void async_copy_b128(const void* g, void* l) {
#ifdef HAVE_ASYNC_LDS
    __builtin_amdgcn_global_load_async_to_lds_b128(
        (gv4i_t*)(uintptr_t)g, (lv4i_t*)(uint32_t)(uintptr_t)l, 0, 0);
#else
    (void)g; (void)l;
#endif
}

template <int N>
__device__ __forceinline__ void wait_asynccnt() {
#if defined(__has_builtin) && __has_builtin(__builtin_amdgcn_s_wait_asynccnt)
    __builtin_amdgcn_s_wait_asynccnt(N);
#else
    asm volatile("s_wait_asynccnt %0" ::"n"(N));
#endif
}

__global__ __launch_bounds__(256) void qconv1x1_wmma_f32(
    const float* __restrict__ x,        // [B, CIN, L]
    const int*   __restrict__ qw,       // [COUT, CIN]
    const float* __restrict__ scale_p,
    const float* __restrict__ wmin_p,
    const int*   __restrict__ qb,       // [COUT]
    const float* __restrict__ bscale_p,
    const float* __restrict__ bmin_p,
    float*       __restrict__ out,      // [B, COUT, L]
    int L)
{
    __shared__ float ldsW[COUT * WSTRIDE];
    __shared__ float ldsX[2][CIN * XSTRIDE];
    __shared__ float ldsBias[COUT];

    const int tid  = threadIdx.x;
    const int lane = tid & 31;
    const int wv   = tid >> 5;
    const int b    = blockIdx.y;
    const int l0   = blockIdx.x * (NTILE * TPB);

    // ---- dequantize weights + bias into LDS (16 KB, L2-resident source) ----
    {
        const float scale = *scale_p;
        const float wmin  = *wmin_p;
        #pragma unroll
        for (int idx = tid; idx < COUT * CIN; idx += 256) {
            const int o = idx >> 6;
            const int i = idx & 63;
            ldsW[o * WSTRIDE + i] = ((float)qw[idx] + QOFF) * scale + wmin;
        }
        if (tid < COUT)
            ldsBias[tid] = ((float)qb[tid] + QOFF) * (*bscale_p) + (*bmin_p);
    }

    const float* xb = x + (size_t)b * CIN * L;

    // stage one 64x128 tile into buffer `bufi`: 8 rows/wave, 1 b128 per lane/row
    auto stage = [&](int bufi, int lt) {
        #pragma unroll
        for (int r = 0; r < 8; ++r) {
            const int row = wv + r * 8;
            const float* g = xb + (size_t)row * L + lt + lane * 4;
            float*       l = &ldsX[bufi][row * XSTRIDE + lane * 4];
#ifdef HAVE_ASYNC_LDS
            async_copy_b128(g, l);          // global -> LDS, no VGPR round-trip
#else
            *(f32x4*)l = *(const f32x4*)g;  // sync fallback
#endif
        }
    };

    stage(0, l0);   // prologue fill

    const int n0   = wv * 16;     // this wave's N sub-tile
    const int lm   = lane & 15;
    const int half = lane >> 4;   // 0: K={0,1}, 1: K={2,3} halves of A/B fragments

    for (int t = 0; t < TPB; ++t) {
        // issue next tile's fill into the other buffer
        if (t + 1 < TPB) stage((t + 1) & 1, l0 + (t + 1) * NTILE);
#ifdef HAVE_ASYNC_LDS
        // async loads complete in order: <=8 outstanding ==> current buffer done
        if (t + 1 < TPB) wait_asynccnt<8>(); else wait_asynccnt<0>();
#endif
        __syncthreads();   // all waves' fills for buffer t&1 visible

        const float* xt = ldsX[t & 1];
        v8f acc[4] = {v8f{}, v8f{}, v8f{}, v8f{}};

        #pragma unroll
        for (int kk = 0; kk < CIN; kk += 4) {
            v2f bf;   // B fragment (4x16 of x)
            {
                const float* bp = &xt[(kk + 2 * half) * XSTRIDE + n0 + lm];
                bf.x = bp[0];
                bf.y = bp[XSTRIDE];
            }
            #pragma unroll
            for (int m = 0; m < 4; ++m) {
                v2f af = *(const v2f*)&ldsW[(m * 16 + lm) * WSTRIDE + kk + 2 * half];
                acc[m] = __builtin_amdgcn_wmma_f32_16x16x4_f32(
                    false, af, false, bf, (short)0, acc[m], false, false);
            }
        }

        // epilogue: bias + non-temporal store (streamed-once output)
        float* ob = out + (size_t)b * COUT * L + (l0 + t * NTILE) + n0 + lm;
        #pragma unroll
        for (int m = 0; m < 4; ++m) {
            #pragma unroll
            for (int r = 0; r < 8; ++r) {
                const int o = m * 16 + half * 8 + r;
                __builtin_nontemporal_store(acc[m][r] + ldsBias[o], ob + (size_t)o * L);
            }
        }

        __syncthreads();   // buffer t&1 free to be overwritten next iteration
    }
}

extern "C" void kernel_launch(void* const* d_in, const int* in_sizes, int n_in,
                              void* d_out, int out_size, void* d_ws, size_t ws_size,
                              hipStream_t stream) {
    const float* x      = (const float*)d_in[0];
    const int*   qw     = (const int*)  d_in[1];
    const float* scale  = (const float*)d_in[2];
    const float* wmin   = (const float*)d_in[3];
    const int*   qb     = (const int*)  d_in[4];
    const float* bscale = (const float*)d_in[5];
    const float* bmin   = (const float*)d_in[6];
    float*       out    = (float*)d_out;

    const int B = 8, Cin = 64;
    const int L = in_sizes[0] / (B * Cin);   // 131072

    dim3 grid(L / (NTILE * TPB), B);
    qconv1x1_wmma_f32<<<grid, 256, 0, stream>>>(x, qw, scale, wmin, qb, bscale, bmin, out, L);
}